// Net4_18519898980804
// MI455X (gfx1250) — compile-verified
//
#include <hip/hip_runtime.h>

#define K_DIM 4096
#define CAP   65536
#define EPSV  1e-8f
#define WAVES_PER_BLOCK 8
#define ROWS_PER_WAVE   16
#define NBLOCKS (CAP / (WAVES_PER_BLOCK * ROWS_PER_WAVE))   // 512

typedef __attribute__((ext_vector_type(2))) float v2f;
typedef __attribute__((ext_vector_type(4))) float v4f;
typedef __attribute__((ext_vector_type(8))) float v8f;

// Monotonic float -> u32 key; pack with ~row so larger key == (larger sim, then smaller row)
__device__ __forceinline__ unsigned long long pack_key(float v, unsigned int row) {
    unsigned int b = __float_as_uint(v);
    b = (b & 0x80000000u) ? ~b : (b | 0x80000000u);
    return ((unsigned long long)b << 32) | (unsigned long long)(~row);
}

// Compute ||x|| (clamped by eps) and reset the packed-argmax atomic slot.
__global__ void prep_kernel(const float* __restrict__ x,
                            float* wsF, unsigned long long* wsP) {
    __shared__ float red[256];
    float s = 0.f;
    for (int i = threadIdx.x; i < K_DIM; i += 256) {
        float v = x[i];
        s = fmaf(v, v, s);
    }
    red[threadIdx.x] = s;
    __syncthreads();
    for (int off = 128; off > 0; off >>= 1) {
        if ((int)threadIdx.x < off) red[threadIdx.x] += red[threadIdx.x + off];
        __syncthreads();
    }
    if (threadIdx.x == 0) {
        wsF[2] = fmaxf(sqrtf(red[0]), EPSV);   // x_norm at ws bytes [8..11]
        wsP[0] = 0ull;                         // packed (sim,row) argmax slot
    }
}

// One wave = 16 rows via V_WMMA_F32_16X16X4_F32.
// A (memory rows) streams from HBM via global_load_b64 (VMEM pipe);
// B (x) is staged in LDS once per block and read via ds_load_b64 (DS pipe).
__global__ void __launch_bounds__(256) gemv_wmma_kernel(
        const float* __restrict__ x, const float* __restrict__ mem,
        float* __restrict__ out, const float* __restrict__ wsF,
        unsigned long long* wsP) {
    __shared__ float              sx[K_DIM];                            // 16 KB staged x
    __shared__ float              snorm[WAVES_PER_BLOCK * ROWS_PER_WAVE];
    __shared__ unsigned long long sbest[WAVES_PER_BLOCK];

    const int tid  = threadIdx.x;
    const int wave = tid >> 5;
    const int lane = tid & 31;
    const int hf   = lane >> 4;      // lane half: supplies K offsets {2h, 2h+1}
    const int m    = lane & 15;      // row within the wave's 16-row tile
    const int rowBase = (blockIdx.x * WAVES_PER_BLOCK + wave) * ROWS_PER_WAVE;

    // Zero the output (uniform per block: blocks [0,256) exactly cover CAP).
    {
        unsigned int gid = blockIdx.x * blockDim.x + tid;
        if (gid < CAP) out[gid] = 0.f;
    }

    // Stage x into LDS: 256 threads x 4 iters x float4 = 4096 floats.
    #pragma unroll
    for (int i = 0; i < 4; ++i) {
        int idx = (i * 256 + tid) * 4;
        *(v4f*)&sx[idx] = *(const v4f*)&x[idx];
    }
    __syncthreads();

    // A layout (16x4 f32): VGPR0 = K {0 | 2}, VGPR1 = K {1 | 3} across lane halves.
    // B layout (4x16 f32): VGPR0 = K {0 | 2}, VGPR1 = K {1 | 3}; all N columns = x[k].
    const float* rowPtr = mem + (size_t)(rowBase + m) * K_DIM + 2 * hf;
    const float* sxPtr  = sx + 2 * hf;

    v8f   c  = {};
    float sq = 0.f;

    #pragma unroll 8
    for (int k = 0; k < K_DIM; k += 4) {
        v2f a = *(const v2f*)(rowPtr + k);          // global_load_b64 (HBM stream)
        v2f b = *(const v2f*)(sxPtr + k);           // ds_load_b64 (LDS, imm offsets)
        c = __builtin_amdgcn_wmma_f32_16x16x4_f32(false, a, false, b,
                                                  (short)0, c, false, false);
        sq = fmaf(a.x, a.x, sq);
        sq = fmaf(a.y, a.y, sq);
    }

    // Combine lane halves: lane L now holds ||row (base + L%16)||^2.
    float n2 = sq + __shfl_xor(sq, 16, 32);
    if (hf == 0) snorm[wave * ROWS_PER_WAVE + m] = n2;
    __syncthreads();

    // D layout: VGPR j = row (j + 8*half); all 16 N columns identical.
    const float xn = wsF[2];
    unsigned long long best = 0ull;
    #pragma unroll
    for (int j = 0; j < 8; ++j) {
        int   r   = j + 8 * hf;
        float mn  = fmaxf(sqrtf(snorm[wave * ROWS_PER_WAVE + r]), EPSV);
        float sim = c[j] / (mn * xn);
        unsigned long long key = pack_key(sim, (unsigned int)(rowBase + r));
        best = (key > best) ? key : best;
    }
    {
        unsigned long long o = __shfl_xor(best, 16, 32);
        best = (o > best) ? o : best;
    }
    if (lane == 0) sbest[wave] = best;
    __syncthreads();
    if (tid == 0) {
        unsigned long long b0 = sbest[0];
        #pragma unroll
        for (int w = 1; w < WAVES_PER_BLOCK; ++w)
            if (sbest[w] > b0) b0 = sbest[w];
        atomicMax(wsP, b0);   // 1 atomic per block (512 total)
    }
}

__global__ void finalize_kernel(const unsigned long long* __restrict__ wsP,
                                float* __restrict__ out) {
    unsigned long long p = wsP[0];
    unsigned int row = ~(unsigned int)(p & 0xffffffffull);
    unsigned int u   = (unsigned int)(p >> 32);
    unsigned int b   = (u & 0x80000000u) ? (u & 0x7fffffffu) : ~u;
    if (row < CAP) out[row] = __uint_as_float(b);
}

extern "C" void kernel_launch(void* const* d_in, const int* in_sizes, int n_in,
                              void* d_out, int out_size, void* d_ws, size_t ws_size,
                              hipStream_t stream) {
    (void)in_sizes; (void)n_in; (void)out_size; (void)ws_size;
    const float* x   = (const float*)d_in[0];   // [4096]
    const float* mem = (const float*)d_in[1];   // [65536, 4096]
    float* out = (float*)d_out;                 // [65536]
    unsigned long long* wsP = (unsigned long long*)d_ws;   // bytes [0..7]
    float*              wsF = (float*)d_ws;                // bytes [8..11] = x_norm

    prep_kernel<<<1, 256, 0, stream>>>(x, wsF, wsP);
    gemv_wmma_kernel<<<NBLOCKS, 256, 0, stream>>>(x, mem, out, wsF, wsP);
    finalize_kernel<<<1, 1, 0, stream>>>(wsP, out);
}